// ECELoss_18202071400747
// MI455X (gfx1250) — compile-verified
//
#include <hip/hip_runtime.h>
#include <stdint.h>

#define NBINS 15

// ---------------------------------------------------------------------------
// ECE loss, MI455X (gfx1250).
// Bandwidth-bound: 524MB logits streamed once (~22.5us @ 23.3TB/s floor).
// Data path: global_load_async_to_lds_b128 (ASYNCcnt, CDNA5 async-tensor
// path), double-buffered per wave; one wave32 per row; ds reads + v_exp_f32.
// No WMMA: the op has no GEMM structure (15-bin histogram beats a mask@vec).
// ---------------------------------------------------------------------------

__global__ void ece_init_kernel(float* __restrict__ bins) {
    int i = threadIdx.x;
    if (i < 3 * NBINS) bins[i] = 0.0f;
}

// One wave per row. C must be a multiple of 4 and <= 1024 (C=1000 here),
// so each row is <= 256 float4 => exactly ceil(C/128)=8 async b128 issues
// per row per wave (the s_wait_asynccnt immediate below assumes 8).
__global__ void __launch_bounds__(128)
ece_main_kernel(const int* __restrict__ labels,
                const float* __restrict__ logits,
                float* __restrict__ gbins,   // [0..14]=cnt [15..29]=sumconf [30..44]=sumacc
                int N, int C) {
    constexpr int WAVES = 4;          // waves per block
    constexpr int BUF_F = 1024;       // floats per buffer (4KB, >= C)
    __shared__ float smem[WAVES * 2 * BUF_F];
    __shared__ float sbins[3 * NBINS];

    const int tid  = threadIdx.x;
    const int lane = tid & 31;
    const int wave = tid >> 5;

    for (int i = tid; i < 3 * NBINS; i += blockDim.x) sbins[i] = 0.0f;
    __syncthreads();

    const int nvec       = C >> 2;                   // float4 per row (250)
    const int totalWaves = gridDim.x * WAVES;
    const int gwave      = blockIdx.x * WAVES + wave;

    float* buf0 = &smem[(wave * 2 + 0) * BUF_F];
    float* buf1 = &smem[(wave * 2 + 1) * BUF_F];

    // Issue the async copy of one row (exactly ceil(nvec/32) instructions,
    // each moving 32 lanes x 16B = 512B; last one EXEC-masked).
    auto issue_row = [&](int row, float* buf) {
        uint64_t g = (uint64_t)(uintptr_t)(logits + (size_t)row * C)
                     + (uint32_t)(lane * 16);
        // Low 32 bits of a flat pointer into LDS == workgroup-relative LDS
        // byte offset (ISA 10.2 aperture mapping) -> what VDST wants.
        uint32_t l = (uint32_t)(uintptr_t)buf + (uint32_t)(lane * 16);
        for (int base = 0; base < nvec; base += 32) {
            if (base + lane < nvec) {
                asm volatile("global_load_async_to_lds_b128 %0, %1, off"
                             :: "v"(l), "v"(g) : "memory");
            }
            g += 512u;
            l += 512u;
        }
    };

    int row = gwave;
    if (row < N) issue_row(row, buf0);
    float* cur = buf0;
    float* nxt = buf1;

    for (; row < N; row += totalWaves) {
        const int nextRow = row + totalWaves;
        if (nextRow < N) {
            issue_row(nextRow, nxt);
            // 8 outstanding from `cur` + 8 just issued for `nxt`;
            // async loads complete in order, so <=8 means `cur` is resident.
            asm volatile("s_wait_asynccnt 8" ::: "memory");
        } else {
            asm volatile("s_wait_asynccnt 0" ::: "memory");
        }

        // ---- pass 1: load row from LDS, per-lane max/argmax --------------
        const float4* s4 = (const float4*)cur;
        float4 v[8];
#pragma unroll
        for (int i = 0; i < 8; ++i) {
            const int idx = lane + 32 * i;
            if (idx < nvec) v[i] = s4[idx];
            else            v[i] = make_float4(-INFINITY, -INFINITY,
                                               -INFINITY, -INFINITY);
        }

        float m  = -INFINITY;
        int   am = 2147483647;
#pragma unroll
        for (int i = 0; i < 8; ++i) {
            const int col0 = (lane + 32 * i) * 4;
            float x;
            x = v[i].x; if (x > m) { m = x; am = col0 + 0; }
            x = v[i].y; if (x > m) { m = x; am = col0 + 1; }
            x = v[i].z; if (x > m) { m = x; am = col0 + 2; }
            x = v[i].w; if (x > m) { m = x; am = col0 + 3; }
        }
        // wave32 butterfly; ties -> smaller column (argmax first-occurrence)
#pragma unroll
        for (int off = 16; off > 0; off >>= 1) {
            const float om = __shfl_xor(m, off, 32);
            const int   oa = __shfl_xor(am, off, 32);
            if (om > m || (om == m && oa < am)) { m = om; am = oa; }
        }

        // ---- pass 2: sum exp(x - max); confidence = 1/sum ----------------
        float s = 0.0f;
#pragma unroll
        for (int i = 0; i < 8; ++i) {
            s += __expf(v[i].x - m) + __expf(v[i].y - m)
               + __expf(v[i].z - m) + __expf(v[i].w - m);
        }
#pragma unroll
        for (int off = 16; off > 0; off >>= 1) s += __shfl_xor(s, off, 32);

        if (lane == 0) {
            const float conf = 1.0f / s;
            const float acc  = (am == labels[row]) ? 1.0f : 0.0f;
            int b = -1;
#pragma unroll
            for (int i = 0; i < NBINS; ++i) {
                const float lo = (float)i       / 15.0f;
                const float hi = (float)(i + 1) / 15.0f;
                if (conf > lo && conf <= hi) b = i;
            }
            if (b >= 0) {
                atomicAdd(&sbins[b],             1.0f);
                atomicAdd(&sbins[NBINS + b],     conf);
                atomicAdd(&sbins[2 * NBINS + b], acc);
            }
        }

        float* t = cur; cur = nxt; nxt = t;
    }

    __syncthreads();
    for (int i = tid; i < 3 * NBINS; i += blockDim.x)
        atomicAdd(&gbins[i], sbins[i]);
}

__global__ void ece_final_kernel(const float* __restrict__ bins,
                                 float* __restrict__ out, float n) {
    if (threadIdx.x == 0) {
        float ece = 0.0f;
        for (int i = 0; i < NBINS; ++i) {
            const float cnt  = bins[i];
            const float sc   = bins[NBINS + i];
            const float sa   = bins[2 * NBINS + i];
            const float safe = fmaxf(cnt, 1.0f);
            const float gap  = fabsf(sc / safe - sa / safe);
            if (cnt > 0.0f) ece += gap * (cnt / n);
        }
        out[0] = 100.0f * ece;
    }
}

extern "C" void kernel_launch(void* const* d_in, const int* in_sizes, int n_in,
                              void* d_out, int out_size, void* d_ws, size_t ws_size,
                              hipStream_t stream) {
    const int*   labels = (const int*)d_in[0];
    const float* logits = (const float*)d_in[1];
    const int N = in_sizes[0];
    const int C = in_sizes[1] / in_sizes[0];   // 1000

    float* gbins = (float*)d_ws;               // 45 floats of scratch

    ece_init_kernel<<<1, 64, 0, stream>>>(gbins);

    constexpr int WAVES_PER_BLOCK = 4;
    constexpr int ROWS_PER_WAVE   = 8;
    const int totalWaves = (N + ROWS_PER_WAVE - 1) / ROWS_PER_WAVE;
    const int blocks     = (totalWaves + WAVES_PER_BLOCK - 1) / WAVES_PER_BLOCK;

    ece_main_kernel<<<blocks, WAVES_PER_BLOCK * 32, 0, stream>>>(
        labels, logits, gbins, N, C);

    ece_final_kernel<<<1, 32, 0, stream>>>(gbins, (float*)d_out, (float)N);
}